// Softmax_12541304504953
// MI455X (gfx1250) — compile-verified
//
#include <hip/hip_runtime.h>
#include <hip/hip_bf16.h>
#include <stdint.h>

#define N_COLS 2048
#define BLOCK 256
#define ROWS_PER_BLOCK 8
#define TILE_BYTES (N_COLS * 4) /* 8192 bytes per row */

typedef float v4f __attribute__((ext_vector_type(4)));

// Low 32 bits of a flat shared-aperture address are the LDS byte offset
// (HW adds LDS_BASE for async-to-LDS VDST addressing, ISA 10/15.18).
__device__ __forceinline__ uint32_t lds_off_of(const void* p) {
    return (uint32_t)(uintptr_t)p;
}

// GLOBAL_LOAD_ASYNC_TO_LDS_B128 (VGLOBAL op 98), GV mode: vdst = LDS byte
// address, vaddr = 64-bit global address, no SADDR.
__device__ __forceinline__ void async_load_b128(uint32_t lds_addr, const void* gaddr) {
    asm volatile("global_load_async_to_lds_b128 %0, %1, off"
                 :
                 : "v"(lds_addr), "v"((unsigned long long)(uintptr_t)gaddr)
                 : "memory");
}

template <int N>
__device__ __forceinline__ void wait_asynccnt() {
#if __has_builtin(__builtin_amdgcn_s_wait_asynccnt)
    __builtin_amdgcn_s_wait_asynccnt(N);
#else
    asm volatile("s_wait_asynccnt %0" ::"n"(N) : "memory");
#endif
    asm volatile("" ::: "memory"); // keep LDS reads below the wait
}

__global__ __launch_bounds__(BLOCK) void softmax_poly_async_kernel(
    const float* __restrict__ x, const float* __restrict__ ec,
    const float* __restrict__ ic, float* __restrict__ out, int rows_total) {
    __shared__ __align__(16) float tile[2][N_COLS]; // double-buffered row tile
    __shared__ float red[BLOCK / 32];

    const int tid  = threadIdx.x;
    const int lane = tid & 31;
    const int wave = tid >> 5;

    const long base_row = (long)blockIdx.x * ROWS_PER_BLOCK;
    if (base_row >= rows_total) return;

    // Uniform coefficient loads -> s_load_b32
    const float e0 = ec[0], e1 = ec[1], e2 = ec[2], e3 = ec[3], e4 = ec[4];
    const float i0 = ic[0], i1 = ic[1], i2 = ic[2], i3 = ic[3], i4 = ic[4];

    // Per-thread byte offsets within a row: two coalesced 16B chunks.
    const uint32_t off0 = (uint32_t)tid * 16u;
    const uint32_t off1 = off0 + (uint32_t)(TILE_BYTES / 2);

    const char* xb = (const char*)x;
    const uint32_t lds_base = lds_off_of(&tile[0][0]);

    // Prime the pipeline: row 0 -> buffer 0 (ASYNCcnt = 2)
    {
        const char* g = xb + (size_t)base_row * TILE_BYTES;
        async_load_b128(lds_base + off0, g + off0);
        async_load_b128(lds_base + off1, g + off1);
    }

    for (int r = 0; r < ROWS_PER_BLOCK; ++r) {
        const uint32_t cur_base = lds_base + (uint32_t)(r & 1) * TILE_BYTES;

        if (r + 1 < ROWS_PER_BLOCK) {
            // Prefetch next row into the other buffer, then wait until only
            // those 2 loads remain outstanding (async loads complete in order).
            const char* g = xb + (size_t)(base_row + r + 1) * TILE_BYTES;
            const uint32_t nxt_base = lds_base + (uint32_t)((r + 1) & 1) * TILE_BYTES;
            async_load_b128(nxt_base + off0, g + off0);
            async_load_b128(nxt_base + off1, g + off1);
            wait_asynccnt<2>();
        } else {
            wait_asynccnt<0>();
        }

        // Each thread reads back exactly the LDS bytes it async-loaded.
        const char* tb = (const char*)&tile[0][0] + (size_t)(r & 1) * TILE_BYTES;
        v4f a = *(const v4f*)(tb + off0);
        v4f b = *(const v4f*)(tb + off1);

        float xs[8] = {a.x, a.y, a.z, a.w, b.x, b.y, b.z, b.w};
        float ev[8];
        float psum = 0.0f;
#pragma unroll
        for (int k = 0; k < 8; ++k) {
            const float xv = xs[k];
            float e = fmaf(e4, xv, e3); // Horner, ascending coeffs
            e = fmaf(e, xv, e2);
            e = fmaf(e, xv, e1);
            e = fmaf(e, xv, e0);
            ev[k] = e;
            psum += e;
        }

        // wave32 shuffle reduction
#pragma unroll
        for (int m = 16; m >= 1; m >>= 1) psum += __shfl_xor(psum, m, 32);
        if (lane == 0) red[wave] = psum;
        __syncthreads();

        float s = 0.0f;
#pragma unroll
        for (int w = 0; w < BLOCK / 32; ++w) s += red[w];

        // 1/s: polynomial seed + 6 Newton-Raphson steps (redundant per thread)
        float inv = fmaf(i4, s, i3);
        inv = fmaf(inv, s, i2);
        inv = fmaf(inv, s, i1);
        inv = fmaf(inv, s, i0);
#pragma unroll
        for (int it = 0; it < 6; ++it) inv = inv * fmaf(-s, inv, 2.0f);

        // Streamed NT stores of e * inv
        char* ob = (char*)(out + (size_t)(base_row + r) * N_COLS);
        v4f oa = {ev[0] * inv, ev[1] * inv, ev[2] * inv, ev[3] * inv};
        v4f obv = {ev[4] * inv, ev[5] * inv, ev[6] * inv, ev[7] * inv};
        __builtin_nontemporal_store(oa, (v4f*)(ob + off0));
        __builtin_nontemporal_store(obv, (v4f*)(ob + off1));

        __syncthreads(); // protect red[] before next row's partials
    }
}

extern "C" void kernel_launch(void* const* d_in, const int* in_sizes, int n_in,
                              void* d_out, int out_size, void* d_ws, size_t ws_size,
                              hipStream_t stream) {
    const float* x  = (const float*)d_in[0];
    const float* ec = (const float*)d_in[1];
    const float* ic = (const float*)d_in[2];
    float* out = (float*)d_out;

    const long total = (long)in_sizes[0];          // 2*16*2048*2048
    const int rows = (int)(total / N_COLS);        // 65536
    const int blocks = rows / ROWS_PER_BLOCK;      // 8192

    softmax_poly_async_kernel<<<dim3(blocks), dim3(BLOCK), 0, stream>>>(
        x, ec, ic, out, rows);
}